// EventFirmGatedGNN_6158983102955
// MI455X (gfx1250) — compile-verified
//
#include <hip/hip_runtime.h>

#define NE 200000
#define NF 50000
#define E_RE 400000
#define E_EF 400000
#define E_FF 800000

typedef float v2f __attribute__((ext_vector_type(2)));
typedef float v8f __attribute__((ext_vector_type(8)));

__device__ __forceinline__ float wave_sum(float v) {
#pragma unroll
  for (int m = 16; m >= 1; m >>= 1) v += __shfl_xor(v, m, 32);
  return v;
}

// ---------------------------------------------------------------------------
// C[N,OutD] = act(A[N,K] @ W[K,OutD] + bias), all row-major f32.
// Block covers 32 rows x OutD cols: blockDim.x = 32*(OutD/16); each wave owns
// one 16-wide column tile and TWO 16x16 M-tiles (B fragment reused in regs).
// grid.x = ceil(N/32); N is always a multiple of 16, so M-tile 0 is always
// fully valid and only M-tile 1 needs a (wave-uniform) tail guard.
// A 32xK tile staged in LDS with padded stride for conflict-free ds loads.
// ---------------------------------------------------------------------------
template <int K, int OutD, int RELU>
__launch_bounds__(256)
__global__ void gemm_wmma_f32(const float* __restrict__ A,
                              const float* __restrict__ W,
                              const float* __restrict__ bias,
                              float* __restrict__ C,
                              int N) {
  constexpr int KP = K + 4;  // pad: lanes hit distinct LDS banks
  __shared__ float As[32 * KP];
  const int m0 = blockIdx.x * 32;
  const int tid = threadIdx.x;
  for (int idx = tid; idx < 32 * K; idx += blockDim.x) {
    int r = idx / K, c = idx - r * K;
    int row = m0 + r;
    As[r * KP + c] = (row < N) ? A[(size_t)row * K + c] : 0.0f;
  }
  __syncthreads();

  const int lane = tid & 31;
  const int n0 = (tid >> 5) * 16;       // wave -> column tile
  const int mi = lane & 15;
  const int n = n0 + mi;
  const int kh = (lane >> 4) << 1;      // upper half-wave holds K+2,K+3

  const float* Wp = W + (size_t)kh * OutD + n;  // immediate offsets: k*OutD*4
  v8f acc0 = {}, acc1 = {};
#pragma unroll
  for (int k = 0; k < K; k += 4) {
    v2f b, a0, a1;
    b.x = Wp[(size_t)k * OutD];
    b.y = Wp[(size_t)(k + 1) * OutD];
    a0.x = As[mi * KP + k + kh];
    a0.y = As[mi * KP + k + kh + 1];
    a1.x = As[(mi + 16) * KP + k + kh];
    a1.y = As[(mi + 16) * KP + k + kh + 1];
    acc0 = __builtin_amdgcn_wmma_f32_16x16x4_f32(false, a0, false, b, (short)0,
                                                 acc0, false, false);
    acc1 = __builtin_amdgcn_wmma_f32_16x16x4_f32(false, a1, false, b, (short)0,
                                                 acc1, false, false);
  }

  const float bn = bias[n];
  // C layout: VGPR i -> row (i + 8*(lane>>4)) of the tile.
  float* Cp = C + (size_t)(m0 + ((lane >> 4) << 3)) * OutD + n;
#pragma unroll
  for (int i = 0; i < 8; ++i) {
    float v = acc0[i] + bn;
    if (RELU) v = fmaxf(v, 0.0f);
    Cp[(size_t)i * OutD] = v;
  }
  if (m0 + 16 < N) {  // wave-uniform tail guard for second M-tile
    float* Cp1 = Cp + (size_t)16 * OutD;
#pragma unroll
    for (int i = 0; i < 8; ++i) {
      float v = acc1[i] + bn;
      if (RELU) v = fmaxf(v, 0.0f);
      Cp1[(size_t)i * OutD] = v;
    }
  }
}

// ---------------------------------------------------------------------------
__global__ void zero_f32(float* __restrict__ p, long n) {
  long i = (long)blockIdx.x * blockDim.x + threadIdx.x;
  long stride = (long)gridDim.x * blockDim.x;
  for (; i < n; i += stride) p[i] = 0.0f;
}

// One wave per edge: acc[dst] += ea * H[src]; cnt[dst] += 1.
__launch_bounds__(256)
__global__ void edge_scatter(const float* __restrict__ H,
                             const int* __restrict__ src,
                             const int* __restrict__ dst,
                             const float* __restrict__ ea,
                             float* __restrict__ acc,
                             float* __restrict__ cnt,
                             int E) {
  int e = blockIdx.x * (blockDim.x >> 5) + (threadIdx.x >> 5);
  if (e >= E) return;
  int lane = threadIdx.x & 31;
  int s = src[e], d = dst[e];
  float w = ea[e];
  const float* hrow = H + (size_t)s * 128;
  float* arow = acc + (size_t)d * 128;
#pragma unroll
  for (int j = 0; j < 4; ++j)
    atomicAdd(arow + lane + 32 * j, w * hrow[lane + 32 * j]);
  if (lane == 0) atomicAdd(cnt + d, 1.0f);
}

// Per-event epilogue: mean-agg, LayerNorm, gate sigmoid, mix, head.
__launch_bounds__(256)
__global__ void event_final(const float* __restrict__ acc, const float* __restrict__ cnt,
                            const float* __restrict__ hself, const float* __restrict__ tg,
                            const float* __restrict__ lng, const float* __restrict__ lnb,
                            const float* __restrict__ gW2, const float* __restrict__ gb2,
                            const float* __restrict__ headW, const float* __restrict__ headb,
                            float* __restrict__ y, float* __restrict__ alpha_o,
                            float* __restrict__ hmix, int N) {
  int i = blockIdx.x * (blockDim.x >> 5) + (threadIdx.x >> 5);
  if (i >= N) return;
  int lane = threadIdx.x & 31;
  float inv = 1.0f / fmaxf(cnt[i], 1.0f);
  float x[4];
  float s = 0.0f;
#pragma unroll
  for (int j = 0; j < 4; ++j) {
    x[j] = acc[(size_t)i * 128 + lane + 32 * j] * inv;
    s += x[j];
  }
  float mu = wave_sum(s) * (1.0f / 128.0f);
  float v = 0.0f;
#pragma unroll
  for (int j = 0; j < 4; ++j) { float d = x[j] - mu; v += d * d; }
  float rstd = rsqrtf(wave_sum(v) * (1.0f / 128.0f) + 1e-5f);

  float z = wave_sum(tg[(size_t)i * 32 + lane] * gW2[lane]) + gb2[0];
  float alpha = 1.0f / (1.0f + __expf(-z));

  float ysum = 0.0f;
#pragma unroll
  for (int j = 0; j < 4; ++j) {
    int k = lane + 32 * j;
    float ln = (x[j] - mu) * rstd * lng[k] + lnb[k];
    float hs = hself[(size_t)i * 128 + k];
    float mix = alpha * hs + (1.0f - alpha) * ln;
    hmix[(size_t)i * 128 + k] = mix;
    ysum += fmaxf(mix, 0.0f) * headW[k];
  }
  ysum = wave_sum(ysum);
  if (lane == 0) {
    y[i] = ysum + headb[0];
    alpha_o[i] = alpha;
  }
}

// Per-firm epilogue: sum of the two mean-aggregations, LayerNorm, + self.
__launch_bounds__(256)
__global__ void firm_final(const float* __restrict__ acc_ef, const float* __restrict__ cnt_ef,
                           const float* __restrict__ acc_ff, const float* __restrict__ cnt_ff,
                           const float* __restrict__ hself,
                           const float* __restrict__ lng, const float* __restrict__ lnb,
                           float* __restrict__ out, int N) {
  int i = blockIdx.x * (blockDim.x >> 5) + (threadIdx.x >> 5);
  if (i >= N) return;
  int lane = threadIdx.x & 31;
  float ie = 1.0f / fmaxf(cnt_ef[i], 1.0f);
  float iff = 1.0f / fmaxf(cnt_ff[i], 1.0f);
  float x[4];
  float s = 0.0f;
#pragma unroll
  for (int j = 0; j < 4; ++j) {
    int k = lane + 32 * j;
    x[j] = acc_ef[(size_t)i * 128 + k] * ie + acc_ff[(size_t)i * 128 + k] * iff;
    s += x[j];
  }
  float mu = wave_sum(s) * (1.0f / 128.0f);
  float v = 0.0f;
#pragma unroll
  for (int j = 0; j < 4; ++j) { float d = x[j] - mu; v += d * d; }
  float rstd = rsqrtf(wave_sum(v) * (1.0f / 128.0f) + 1e-5f);
#pragma unroll
  for (int j = 0; j < 4; ++j) {
    int k = lane + 32 * j;
    float ln = (x[j] - mu) * rstd * lng[k] + lnb[k];
    out[(size_t)i * 128 + k] = hself[(size_t)i * 128 + k] + ln;
  }
}

// ---------------------------------------------------------------------------
extern "C" void kernel_launch(void* const* d_in, const int* in_sizes, int n_in,
                              void* d_out, int out_size, void* d_ws, size_t ws_size,
                              hipStream_t stream) {
  const float* x_e   = (const float*)d_in[0];
  const float* x_f   = (const float*)d_in[1];
  const int* src_re  = (const int*)d_in[2];
  const int* dst_re  = (const int*)d_in[3];
  const float* ea_re = (const float*)d_in[4];
  const int* src_ef  = (const int*)d_in[5];
  const int* dst_ef  = (const int*)d_in[6];
  const float* ea_ef = (const float*)d_in[7];
  const int* src_ff  = (const int*)d_in[8];
  const int* dst_ff  = (const int*)d_in[9];
  const float* ea_ff = (const float*)d_in[10];
  const float* ev_W1 = (const float*)d_in[11];
  const float* ev_b1 = (const float*)d_in[12];
  const float* ev_W2 = (const float*)d_in[13];
  const float* ev_b2 = (const float*)d_in[14];
  const float* fm_W1 = (const float*)d_in[15];
  const float* fm_b1 = (const float*)d_in[16];
  const float* fm_W2 = (const float*)d_in[17];
  const float* fm_b2 = (const float*)d_in[18];
  const float* W_re  = (const float*)d_in[19];
  const float* b_re  = (const float*)d_in[20];
  const float* W_ef  = (const float*)d_in[21];
  const float* b_ef  = (const float*)d_in[22];
  const float* W_ff  = (const float*)d_in[23];
  const float* b_ff  = (const float*)d_in[24];
  const float* ln_e_g = (const float*)d_in[25];
  const float* ln_e_b = (const float*)d_in[26];
  const float* ln_f_g = (const float*)d_in[27];
  const float* ln_f_b = (const float*)d_in[28];
  const float* g_W1  = (const float*)d_in[29];
  const float* g_b1  = (const float*)d_in[30];
  const float* g_W2  = (const float*)d_in[31];
  const float* g_b2  = (const float*)d_in[32];
  const float* head_W = (const float*)d_in[33];
  const float* head_b = (const float*)d_in[34];

  float* ws = (float*)d_ws;
  size_t o = 0;
  float* acc_re = ws + o; o += (size_t)NE * 128;
  float* acc_ef = ws + o; o += (size_t)NF * 128;
  float* acc_ff = ws + o; o += (size_t)NF * 128;
  float* cnt_re = ws + o; o += NE;
  float* cnt_ef = ws + o; o += NF;
  float* cnt_ff = ws + o; o += NF;
  size_t zero_n = o;  // everything above must be zeroed each call
  float* t_e      = ws + o; o += (size_t)NE * 128;  // later reused as hef
  float* h_self_e = ws + o; o += (size_t)NE * 128;
  float* t_f      = ws + o; o += (size_t)NF * 128;  // later reused as hre
  float* h_self_f = ws + o; o += (size_t)NF * 128;
  float* hff      = ws + o; o += (size_t)NF * 128;
  float* t_g      = ws + o; o += (size_t)NE * 32;

  float* y_out     = (float*)d_out;
  float* alpha_out = y_out + NE;
  float* hmix_out  = alpha_out + NE;
  float* firm_out  = hmix_out + (size_t)NE * 128;

  const int gE = (NE + 31) / 32;  // 6250
  const int gF = (NF + 31) / 32;  // 1563 (tail block: only M-tile 0 valid)

  zero_f32<<<2048, 256, 0, stream>>>(ws, (long)zero_n);

  // Self-path MLP encoders (ReLU fused into GEMM epilogue).
  gemm_wmma_f32<64, 128, 1><<<gE, 256, 0, stream>>>(x_e, ev_W1, ev_b1, t_e, NE);
  gemm_wmma_f32<128, 128, 1><<<gE, 256, 0, stream>>>(t_e, ev_W2, ev_b2, h_self_e, NE);
  gemm_wmma_f32<128, 128, 1><<<gF, 256, 0, stream>>>(x_f, fm_W1, fm_b1, t_f, NF);
  gemm_wmma_f32<128, 128, 1><<<gF, 256, 0, stream>>>(t_f, fm_W2, fm_b2, h_self_f, NF);

  // Relation linears (reuse the temp buffers: t_f -> hre, t_e -> hef).
  float* hre = t_f;
  float* hef = t_e;
  gemm_wmma_f32<128, 128, 0><<<gF, 256, 0, stream>>>(h_self_f, W_re, b_re, hre, NF);
  gemm_wmma_f32<128, 128, 0><<<gE, 256, 0, stream>>>(h_self_e, W_ef, b_ef, hef, NE);
  gemm_wmma_f32<128, 128, 0><<<gF, 256, 0, stream>>>(h_self_f, W_ff, b_ff, hff, NF);

  // Gate hidden: [NE,128] x [128,32], 2 column tiles -> 64-thread blocks.
  gemm_wmma_f32<128, 32, 1><<<gE, 64, 0, stream>>>(h_self_e, g_W1, g_b1, t_g, NE);

  // Edge scatters (one wave per edge, f32 atomics into L2).
  edge_scatter<<<(E_RE + 7) / 8, 256, 0, stream>>>(hre, src_re, dst_re, ea_re, acc_re, cnt_re, E_RE);
  edge_scatter<<<(E_EF + 7) / 8, 256, 0, stream>>>(hef, src_ef, dst_ef, ea_ef, acc_ef, cnt_ef, E_EF);
  edge_scatter<<<(E_FF + 7) / 8, 256, 0, stream>>>(hff, src_ff, dst_ff, ea_ff, acc_ff, cnt_ff, E_FF);

  // Fused epilogues.
  event_final<<<NE / 8, 256, 0, stream>>>(acc_re, cnt_re, h_self_e, t_g,
                                          ln_e_g, ln_e_b, g_W2, g_b2, head_W, head_b,
                                          y_out, alpha_out, hmix_out, NE);
  firm_final<<<NF / 8, 256, 0, stream>>>(acc_ef, cnt_ef, acc_ff, cnt_ff, h_self_f,
                                         ln_f_g, ln_f_b, firm_out, NF);
}